// _ProposalLayer_33097017982976
// MI455X (gfx1250) — compile-verified
//
#include <hip/hip_runtime.h>
#include <stdint.h>

// ---------------- problem constants (from reference) ----------------
#define BATCH   32
#define AANCH   9
#define FH_     64
#define FW_     96
#define NPROP   (FH_*FW_*AANCH)   // 55296 proposals per batch
#define PADN    65536             // next pow2 for bitonic sort
#define PRE     6000              // PRE_NMS_TOP_N
#define POST    300               // POST_NMS_TOP_N
#define NMS_T   0.7f
#define TILE    4096              // LDS bitonic tile (32KB of u64 keys)
#define NMS_THREADS 256
#define NWORDS  ((PRE + 31) / 32) // 188 valid-mask words

// anchor widths/heights, ravel order = ratio-major, scale-minor
// ratios {0.5,1,2} -> ws_r {23,16,11}, hs_r {12,16,22}; scales {8,16,32}
__constant__ float WSA[9] = {184.f,368.f,736.f, 128.f,256.f,512.f,  88.f,176.f,352.f};
__constant__ float HSA[9] = { 96.f,192.f,384.f, 128.f,256.f,512.f, 176.f,352.f,704.f};

// ---------------- CDNA5 async global<->LDS helpers ----------------
#define GLOBAL_AS __attribute__((address_space(1)))
#define LDS_AS    __attribute__((address_space(3)))

typedef __attribute__((ext_vector_type(4))) int v4i;

#if defined(__has_builtin)
#if __has_builtin(__builtin_amdgcn_global_load_async_to_lds_b128)
#define HAVE_ASYNC_LOAD_B128 1
#endif
#if __has_builtin(__builtin_amdgcn_global_store_async_from_lds_b128)
#define HAVE_ASYNC_STORE_B128 1
#endif
#if __has_builtin(__builtin_amdgcn_s_wait_asynccnt)
#define HAVE_WAIT_ASYNC 1
#endif
#endif

__device__ __forceinline__ void async_copy16(void* lds_dst, const void* gsrc) {
#if defined(HAVE_ASYNC_LOAD_B128)
  __builtin_amdgcn_global_load_async_to_lds_b128(
      (GLOBAL_AS v4i*)(uintptr_t)gsrc,
      (LDS_AS   v4i*)(unsigned)(uintptr_t)lds_dst,
      0, 0);
#else
  // ISA: VDST = 32-bit LDS byte address, VADDR = 64-bit global address
  asm volatile("global_load_async_to_lds_b128 %0, %1, off"
               :: "v"((unsigned)(uintptr_t)lds_dst), "v"(gsrc)
               : "memory");
#endif
}

__device__ __forceinline__ void async_store16(void* gdst, const void* lds_src) {
#if defined(HAVE_ASYNC_STORE_B128)
  __builtin_amdgcn_global_store_async_from_lds_b128(
      (GLOBAL_AS v4i*)(uintptr_t)gdst,
      (LDS_AS   v4i*)(unsigned)(uintptr_t)lds_src,
      0, 0);
#else
  // ISA: VADDR = 64-bit global address, VSRC = 32-bit LDS byte address
  asm volatile("global_store_async_from_lds_b128 %0, %1, off"
               :: "v"(gdst), "v"((unsigned)(uintptr_t)lds_src)
               : "memory");
#endif
}

__device__ __forceinline__ void async_wait0() {
#if defined(HAVE_WAIT_ASYNC)
  __builtin_amdgcn_s_wait_asynccnt(0);
#else
  asm volatile("s_wait_asynccnt 0" ::: "memory");
#endif
}

// ---------------- stage 1: decode proposals + build sort keys ----------------
__device__ __forceinline__ unsigned fkey(float f) {
  unsigned u = __float_as_uint(f);
  return (u & 0x80000000u) ? ~u : (u | 0x80000000u);
}

__global__ void k_proposals(const float* __restrict__ cls,
                            const float* __restrict__ dlt,
                            const float* __restrict__ iminfo,
                            float4* __restrict__ props,
                            unsigned long long* __restrict__ keys) {
  unsigned flat = blockIdx.x * blockDim.x + threadIdx.x;   // [0, BATCH*PADN)
  unsigned b = flat >> 16;
  unsigned i = flat & 0xFFFFu;
  if (i >= NPROP) { keys[flat] = 0ull; return; }           // padding sinks to bottom

  unsigned a = i % AANCH;
  unsigned t = i / AANCH;
  unsigned x = t % FW_;
  unsigned y = t / FW_;

  // score = rpn_cls_prob[b, A+a, y, x]   (NCHW)
  size_t sidx = (((size_t)b * (2*AANCH) + AANCH + a) * FH_ + y) * FW_ + x;
  float sc = cls[sidx];

  // deltas channel = a*4 + c
  const size_t plane = (size_t)FH_ * FW_;
  size_t dbase = (((size_t)b * (4*AANCH) + 4*a) * FH_ + y) * FW_ + x;
  __builtin_prefetch(dlt + dbase + 2*plane, 0, 1);          // global_prefetch_b8
  float dx = dlt[dbase];
  float dy = dlt[dbase +   plane];
  float dw = dlt[dbase + 2*plane];
  float dh = dlt[dbase + 3*plane];

  float w = WSA[a], h = HSA[a];
  float sx = (float)x * 16.0f, sy = (float)y * 16.0f;
  float x1 = sx + 7.5f - 0.5f * (w - 1.0f);
  float y1 = sy + 7.5f - 0.5f * (h - 1.0f);
  float cx = x1 + 0.5f * w;
  float cy = y1 + 0.5f * h;

  float pcx = dx * w + cx;
  float pcy = dy * h + cy;
  float pw  = expf(dw) * w;
  float ph  = expf(dh) * h;

  float bx1 = pcx - 0.5f * pw;
  float by1 = pcy - 0.5f * ph;
  float bx2 = pcx + 0.5f * pw;
  float by2 = pcy + 0.5f * ph;

  float wmax = iminfo[b*3 + 1] - 1.0f;
  float hmax = iminfo[b*3 + 0] - 1.0f;
  bx1 = fminf(fmaxf(bx1, 0.0f), wmax);
  by1 = fminf(fmaxf(by1, 0.0f), hmax);
  bx2 = fminf(fmaxf(bx2, 0.0f), wmax);
  by2 = fminf(fmaxf(by2, 0.0f), hmax);

  props[(size_t)b * NPROP + i] = make_float4(bx1, by1, bx2, by2);
  // descending sort key: score bits high, ~index low (stable: lower index first)
  keys[flat] = ((unsigned long long)fkey(sc) << 32) | (unsigned long long)(0xFFFFFFFFu - i);
}

// ---------------- stage 2: per-batch bitonic sort (descending) ----------------
__global__ void k_bitonic_global(unsigned long long* __restrict__ keys,
                                 unsigned k, unsigned j) {
  unsigned flat = blockIdx.x * blockDim.x + threadIdx.x;
  unsigned il = flat & 0xFFFFu;            // index within batch
  unsigned ixj = il ^ j;
  if (ixj <= il) return;
  unsigned base = flat & ~0xFFFFu;
  unsigned long long va = keys[base + il];
  unsigned long long vb = keys[base + ixj];
  bool desc = ((il & k) == 0);
  if (desc ? (va < vb) : (va > vb)) {
    keys[base + il]  = vb;
    keys[base + ixj] = va;
  }
}

// LDS-tiled stages: for k in [k_first..k_last], j from jstart (j_first for the
// first k, else k/2) down to 1 — all within a 4096-element tile (32KB LDS).
// Tile fill and write-back use the CDNA5 async LDS path (no VGPR round trip).
__global__ void k_bitonic_local(unsigned long long* __restrict__ keys,
                                unsigned k_first, unsigned k_last, unsigned j_first) {
  __shared__ __align__(16) unsigned long long tile[TILE];
  unsigned base   = blockIdx.x * TILE;       // flat base (tiles never cross batches)
  unsigned ilbase = base & 0xFFFFu;          // within-batch base

  // async fill: 2 keys per b128 transfer, straight into LDS
  for (unsigned e = threadIdx.x * 2u; e < TILE; e += blockDim.x * 2u)
    async_copy16(&tile[e], &keys[base + e]);
  async_wait0();
  __syncthreads();

  for (unsigned k = k_first; k <= k_last; k <<= 1) {
    unsigned jstart = (k == k_first) ? j_first : (k >> 1);
    for (unsigned j = jstart; j >= 1; j >>= 1) {
      for (unsigned p = threadIdx.x; p < TILE/2; p += blockDim.x) {
        unsigned i = ((p & ~(j - 1u)) << 1) | (p & (j - 1u));
        unsigned gi = ilbase + i;
        unsigned long long va = tile[i];
        unsigned long long vb = tile[i + j];
        bool desc = ((gi & k) == 0);
        if (desc ? (va < vb) : (va > vb)) {
          tile[i]     = vb;
          tile[i + j] = va;
        }
      }
      __syncthreads();
    }
  }

  // async write-back straight from LDS (S_ENDPGM implicitly waits all counters)
  for (unsigned e = threadIdx.x * 2u; e < TILE; e += blockDim.x * 2u)
    async_store16(&keys[base + e], &tile[e]);
}

// ---------------- stage 3: gather top-6000 boxes in score order ----------------
__global__ void k_gather(const unsigned long long* __restrict__ keys,
                         const float4* __restrict__ props,
                         float4* __restrict__ sorted) {
  unsigned idx = blockIdx.x * blockDim.x + threadIdx.x;   // [0, BATCH*PRE)
  if (idx >= BATCH * PRE) return;
  unsigned b = idx / PRE;
  unsigned r = idx - b * PRE;
  unsigned long long kk = keys[(size_t)b * PADN + r];
  unsigned i = 0xFFFFFFFFu - (unsigned)(kk & 0xFFFFFFFFull);
  const float4* src = props + (size_t)b * NPROP + i;
  __builtin_prefetch(src, 0, 3);
  sorted[idx] = *src;
}

// ---------------- stage 4: greedy NMS, one block per batch ----------------
// 6000 boxes (96KB) + areas (24KB) live in LDS (CDNA5: 320KB per WGP).
__global__ void __launch_bounds__(NMS_THREADS)
k_nms(const float4* __restrict__ sorted, float* __restrict__ out) {
  extern __shared__ char smem[];
  float4* sbox  = (float4*)smem;                                 // PRE * 16B
  float*  sarea = (float*)(smem + (size_t)PRE * sizeof(float4)); // PRE * 4B
  __shared__ unsigned svalid[NWORDS];
  __shared__ unsigned swmin[NMS_THREADS / 32];
  __shared__ int   s_cur;
  __shared__ float s_bx, s_by, s_bz, s_bw, s_ba;

  const int b = blockIdx.x;
  const float4* src = sorted + (size_t)b * PRE;

  // async stage boxes global -> LDS (ASYNCcnt)
  for (int r = threadIdx.x; r < PRE; r += NMS_THREADS)
    async_copy16(&sbox[r], &src[r]);
  async_wait0();
  __syncthreads();

  for (int r = threadIdx.x; r < PRE; r += NMS_THREADS) {
    float4 v = sbox[r];
    sarea[r] = (v.z - v.x + 1.0f) * (v.w - v.y + 1.0f);
  }
  for (int w = threadIdx.x; w < NWORDS; w += NMS_THREADS) {
    int rem = PRE - w * 32;
    svalid[w] = (rem >= 32) ? 0xFFFFFFFFu : ((1u << rem) - 1u);
  }
  __syncthreads();

  for (int it = 0; it < POST; ++it) {
    // first-valid index == argmax over the boolean valid array (sorted order)
    unsigned cand = 0xFFFFFFFFu;
    if (threadIdx.x < NWORDS) {
      unsigned wv = svalid[threadIdx.x];
      if (wv) cand = threadIdx.x * 32u + (unsigned)(__ffs(wv) - 1);
    }
    // wave32 min-reduce, then across 8 waves via LDS
    unsigned v = cand;
    for (int off = 16; off > 0; off >>= 1) {
      unsigned o = (unsigned)__shfl_xor((int)v, off, 32);
      v = (o < v) ? o : v;
    }
    if ((threadIdx.x & 31) == 0) swmin[threadIdx.x >> 5] = v;
    __syncthreads();
    if (threadIdx.x == 0) {
      unsigned m = swmin[0];
      #pragma unroll
      for (int w = 1; w < NMS_THREADS / 32; ++w) m = (swmin[w] < m) ? swmin[w] : m;
      float* row = out + ((size_t)b * POST + it) * 5;
      row[0] = (float)b;
      if (m != 0xFFFFFFFFu) {
        float4 bb = sbox[m];
        row[1] = bb.x; row[2] = bb.y; row[3] = bb.z; row[4] = bb.w;
        s_cur = (int)m;
        s_bx = bb.x; s_by = bb.y; s_bz = bb.z; s_bw = bb.w;
        s_ba = sarea[m];
      } else {
        row[1] = 0.f; row[2] = 0.f; row[3] = 0.f; row[4] = 0.f;
        s_cur = -1;
      }
    }
    __syncthreads();

    if (s_cur >= 0 && threadIdx.x < NWORDS) {
      unsigned wv = svalid[threadIdx.x];
      if (wv) {
        float cx1 = s_bx, cy1 = s_by, cx2 = s_bz, cy2 = s_bw, ca = s_ba;
        unsigned newv = wv, m = wv;
        while (m) {
          int bit = __ffs(m) - 1;
          m &= m - 1u;
          int e = threadIdx.x * 32 + bit;
          float4 vb = sbox[e];
          float xx1 = fmaxf(cx1, vb.x);
          float yy1 = fmaxf(cy1, vb.y);
          float xx2 = fminf(cx2, vb.z);
          float yy2 = fminf(cy2, vb.w);
          float iw = fmaxf(xx2 - xx1 + 1.0f, 0.0f);
          float ih = fmaxf(yy2 - yy1 + 1.0f, 0.0f);
          float inter = iw * ih;
          float iou = inter / (ca + sarea[e] - inter);
          if (iou > NMS_T) newv &= ~(1u << bit);
        }
        svalid[threadIdx.x] = newv;   // each thread owns its word: no atomics
      }
    }
    __syncthreads();
  }
}

// ---------------- host launch ----------------
extern "C" void kernel_launch(void* const* d_in, const int* in_sizes, int n_in,
                              void* d_out, int out_size, void* d_ws, size_t ws_size,
                              hipStream_t stream) {
  (void)in_sizes; (void)n_in; (void)out_size; (void)ws_size;
  const float* cls = (const float*)d_in[0];   // (32, 18, 64, 96)
  const float* dlt = (const float*)d_in[1];   // (32, 36, 64, 96)
  const float* imi = (const float*)d_in[2];   // (32, 3)
  float* out = (float*)d_out;                 // (32, 300, 5)

  // workspace layout: keys (16MB) | proposals (28.3MB) | sorted boxes (3MB)
  unsigned long long* keys = (unsigned long long*)d_ws;
  float4* props  = (float4*)(keys + (size_t)BATCH * PADN);
  float4* sorted = props + (size_t)BATCH * NPROP;

  const dim3 blk(256);
  const unsigned gridFlat = (BATCH * PADN) / 256;   // 8192
  const unsigned gridTile = (BATCH * PADN) / TILE;  // 512

  k_proposals<<<gridFlat, blk, 0, stream>>>(cls, dlt, imi, props, keys);

  // full bitonic sort per batch: LDS tiles handle j<=2048, global passes j>2048
  k_bitonic_local<<<gridTile, blk, 0, stream>>>(keys, 2u, (unsigned)TILE, 1u);
  for (unsigned k = TILE * 2; k <= PADN; k <<= 1) {
    for (unsigned j = k >> 1; j > TILE / 2; j >>= 1)
      k_bitonic_global<<<gridFlat, blk, 0, stream>>>(keys, k, j);
    k_bitonic_local<<<gridTile, blk, 0, stream>>>(keys, k, k, (unsigned)(TILE / 2));
  }

  k_gather<<<(BATCH * PRE + 255) / 256, blk, 0, stream>>>(keys, props, sorted);

  const size_t nms_lds = (size_t)PRE * sizeof(float4) + (size_t)PRE * sizeof(float);
  k_nms<<<BATCH, NMS_THREADS, nms_lds, stream>>>(sorted, out);
}